// ManualDecodeAttention_43885975830657
// MI455X (gfx1250) — compile-verified
//
#include <hip/hip_runtime.h>

typedef __attribute__((ext_vector_type(2))) float    v2f;
typedef __attribute__((ext_vector_type(8))) float    v8f;
typedef __attribute__((ext_vector_type(4))) unsigned v4u;
typedef __attribute__((ext_vector_type(4))) int      v4i;
typedef __attribute__((ext_vector_type(8))) int      v8i;

#define NUM_HEADS 32
#define NUM_KV    8
#define GROUPS    4
#define HD        128
#define SEQ       4096
#define BATCH     32

#define WAVES     8
#define TILE      16
#define KSTR      132                        // 128 dwords + TDM pad of 4 dwords
#define VSTR      132
#define BUF_FLTS  (TILE*KSTR + TILE*VSTR)    // one K+V tile buffer (floats)
#define WAVE_LDS  (2 * BUF_FLTS)             // double buffered
#define SPLIT     (SEQ / WAVES)              // 512 keys per wave
#define NTILES    (SPLIT / TILE)             // 32 tiles per wave

// Raw 32-bit LDS byte offset of a shared-memory pointer (generic -> AS3).
__device__ static inline unsigned lds_addr_of(const void* p) {
  return (unsigned)(unsigned long long)(__attribute__((address_space(3))) const char*)p;
}

// TDM: DMA one 16-row x 128-dword tile (row stride 1024 dwords in memory)
// into LDS with 4 pad dwords inserted every 128 dwords (LDS row stride 132).
__device__ static inline void tdm_load_tile(unsigned lds_byte, const float* gp) {
  unsigned long long ga = (unsigned long long)(size_t)gp;
  unsigned ga_lo = (unsigned)__builtin_amdgcn_readfirstlane((int)(unsigned)(ga & 0xFFFFFFFFull));
  unsigned ga_hi = (unsigned)__builtin_amdgcn_readfirstlane((int)(unsigned)(ga >> 32));
  unsigned lo    = (unsigned)__builtin_amdgcn_readfirstlane((int)lds_byte);

  v4u g0;
  g0[0] = 1u;                                   // count=1 (valid user descriptor)
  g0[1] = lo;                                   // lds_addr (bytes)
  g0[2] = ga_lo;                                // global_addr[31:0]
  g0[3] = (ga_hi & 0x01FFFFFFu) | (2u << 30);   // global_addr[56:32] | type=2

  v8i g1;
  g1[0] = (int)((2u << 16)      // data_size = 4B
              | (1u << 20)      // pad_enable
              | (6u << 22)      // pad_interval: every 128 dwords
              | (3u << 25));    // pad_amount: 4 dwords  -> LDS stride 132
  g1[1] = (int)(128u << 16);    // tensor_dim0 = 128 elements
  g1[2] = (int)((unsigned)TILE << 16);  // tensor_dim1 = 16 rows
  g1[3] = (int)(128u << 16);    // tile_dim0 = 128
  g1[4] = TILE;                 // tile_dim1 = 16 (tile_dim2 = 0)
  g1[5] = 1024;                 // tensor_dim0_stride = NUM_KV*HD elements
  g1[6] = 0;
  g1[7] = 0;

  v4i z4 = {0, 0, 0, 0};
#if __clang_major__ >= 23
  v8i z8 = {0, 0, 0, 0, 0, 0, 0, 0};
  __builtin_amdgcn_tensor_load_to_lds(g0, g1, z4, z4, z8, 0);
#else
  __builtin_amdgcn_tensor_load_to_lds(g0, g1, z4, z4, 0);
#endif
}

__global__ __launch_bounds__(WAVES * 32) void gqa_decode_wmma(
    const float* __restrict__ q,
    const float* __restrict__ kc,
    const float* __restrict__ vc,
    float* __restrict__ out)
{
  extern __shared__ float smem[];
  const int tid  = threadIdx.x;
  const int wave = tid >> 5;
  const int lane = tid & 31;
  const int n16  = lane & 15;   // N-slot: head for scores/out, M-slot for A operands
  const int half = lane >> 4;   // which K-pair / M-upper-half this lane covers

  const int b  = blockIdx.x >> 3;   // batch
  const int kv = blockIdx.x & 7;    // kv head

  float* wbase = smem + wave * WAVE_LDS;
  const unsigned wave_lds_byte = lds_addr_of(wbase);

  const float scale = 0.08838834764831845f;  // 1/sqrt(128)

  // ---- Load Q^T as WMMA B-operand: lane holds head n16 (valid < 4),
  //      chunk c supplies dims {4c+2*half, 4c+2*half+1}, pre-scaled. ----
  v2f qb[32];
  if (n16 < GROUPS) {
    const float* qp = q + ((size_t)b * NUM_HEADS + kv * GROUPS + n16) * HD;
    #pragma unroll
    for (int c = 0; c < 32; ++c) {
      const int d = 4 * c + 2 * half;
      v2f t; t.x = qp[d] * scale; t.y = qp[d + 1] * scale;
      qb[c] = t;
    }
  } else {
    #pragma unroll
    for (int c = 0; c < 32; ++c) { v2f z; z.x = 0.f; z.y = 0.f; qb[c] = z; }
  }

  // out^T accumulators: 8 dim-tiles of 16; lane n16 = head, vgpr r = dim r+8*half
  v8f acc[8] = {};
  float run_m = -1e30f, run_s = 0.f;

  const size_t krow = (size_t)NUM_KV * HD;  // 1024 floats between consecutive keys
  const float* kg0 = kc + ((size_t)b * SEQ * NUM_KV + kv) * HD + (size_t)(wave * SPLIT) * krow;
  const float* vg0 = vc + ((size_t)b * SEQ * NUM_KV + kv) * HD + (size_t)(wave * SPLIT) * krow;

  // ---- Prologue: TDM-fetch tile 0 into buffer 0 ----
  tdm_load_tile(wave_lds_byte, kg0);
  tdm_load_tile(wave_lds_byte + TILE * KSTR * 4, vg0);

  for (int t = 0; t < NTILES; ++t) {
    const int buf = t & 1;
    float* kbuf = wbase + buf * BUF_FLTS;
    float* vbuf = kbuf + TILE * KSTR;

    // ---- Issue next tile into the other buffer, then wait for this tile.
    //      (TDM ops of a wave complete in order: <=2 outstanding => tile t done.)
    if (t + 1 < NTILES) {
      const unsigned nb = wave_lds_byte + ((t + 1) & 1) * (BUF_FLTS * 4);
      tdm_load_tile(nb, kg0 + (size_t)(t + 1) * TILE * krow);
      tdm_load_tile(nb + TILE * KSTR * 4, vg0 + (size_t)(t + 1) * TILE * krow);
      __builtin_amdgcn_s_wait_tensorcnt(2);
    } else {
      __builtin_amdgcn_s_wait_tensorcnt(0);
    }
    asm volatile("" ::: "memory");  // keep ds_loads below the tensorcnt wait

    // ---- scores^T = K_tile(A: M=key, K=dim) x Q^T(B: N=head) ----
    v8f st = {};
    #pragma unroll
    for (int c = 0; c < 32; ++c) {
      v2f ka = *(const v2f*)(kbuf + n16 * KSTR + 4 * c + 2 * half);
      st = __builtin_amdgcn_wmma_f32_16x16x4_f32(
          false, ka, false, qb[c], (short)0, st, false, false);
    }

    // ---- online softmax over this tile; lane holds 8 of its head's 16 keys ----
    float tmax = -1e30f;
    #pragma unroll
    for (int r = 0; r < 8; ++r) tmax = fmaxf(tmax, st[r]);
    tmax = fmaxf(tmax, __shfl_xor(tmax, 16));
    const float nm    = fmaxf(run_m, tmax);
    const float alpha = __expf(run_m - nm);
    float p[8], tv[8], tsum = 0.f;
    #pragma unroll
    for (int r = 0; r < 8; ++r) { p[r] = __expf(st[r] - nm); tsum += p[r]; }
    tsum += __shfl_xor(tsum, 16);
    run_s = run_s * alpha + tsum;
    run_m = nm;
    #pragma unroll
    for (int dt = 0; dt < 8; ++dt) {
      #pragma unroll
      for (int j = 0; j < 8; ++j) acc[dt][j] *= alpha;
    }

    // ---- Build P^T B-operand: swap key halves across lane halves ----
    #pragma unroll
    for (int r = 0; r < 8; ++r) tv[r] = __shfl_xor(p[r], 16);
    v2f pb[4];
    pb[0].x = half ? tv[2] : p[0];  pb[0].y = half ? tv[3] : p[1];
    pb[1].x = half ? tv[6] : p[4];  pb[1].y = half ? tv[7] : p[5];
    pb[2].x = half ? p[2]  : tv[0]; pb[2].y = half ? p[3]  : tv[1];
    pb[3].x = half ? p[6]  : tv[4]; pb[3].y = half ? p[7]  : tv[5];

    // ---- out^T += V^T(A: M=dim, K=key) x P^T(B: N=head) ----
    #pragma unroll
    for (int c = 0; c < 4; ++c) {
      const float* vr = vbuf + (4 * c + 2 * half) * VSTR + n16;
      #pragma unroll
      for (int dt = 0; dt < 8; ++dt) {
        v2f va; va.x = vr[dt * 16]; va.y = vr[dt * 16 + VSTR];
        acc[dt] = __builtin_amdgcn_wmma_f32_16x16x4_f32(
            false, va, false, pb[c], (short)0, acc[dt], false, false);
      }
    }
  }

  // ---- Cross-wave flash combine via LDS ----
  __syncthreads();  // all TDM done (tensorcnt==0) and staging regions dead; reuse smem
  float* accbuf = smem;                              // WAVES*GROUPS*HD floats
  float* mbuf   = smem + WAVES * GROUPS * HD;        // WAVES*GROUPS
  float* sbuf   = mbuf + WAVES * GROUPS;             // WAVES*GROUPS

  if (n16 < GROUPS) {
    float* ap = accbuf + (wave * GROUPS + n16) * HD;
    #pragma unroll
    for (int dt = 0; dt < 8; ++dt) {
      #pragma unroll
      for (int r = 0; r < 8; ++r)
        ap[dt * 16 + r + 8 * half] = acc[dt][r];
    }
    if (half == 0) {
      mbuf[wave * GROUPS + n16] = run_m;
      sbuf[wave * GROUPS + n16] = run_s;
    }
  }
  __syncthreads();

  for (int i = tid; i < GROUPS * HD; i += WAVES * 32) {
    const int h = i >> 7, d = i & (HD - 1);
    float gm = -1e30f;
    #pragma unroll
    for (int w = 0; w < WAVES; ++w) gm = fmaxf(gm, mbuf[w * GROUPS + h]);
    float num = 0.f, den = 0.f;
    #pragma unroll
    for (int w = 0; w < WAVES; ++w) {
      const float wt = __expf(mbuf[w * GROUPS + h] - gm);
      num += wt * accbuf[(w * GROUPS + h) * HD + d];
      den += wt * sbuf[w * GROUPS + h];
    }
    out[((size_t)b * NUM_HEADS + kv * GROUPS + h) * HD + d] = num / den;
  }
}

extern "C" void kernel_launch(void* const* d_in, const int* in_sizes, int n_in,
                              void* d_out, int out_size, void* d_ws, size_t ws_size,
                              hipStream_t stream) {
  const float* q  = (const float*)d_in[0];
  const float* kc = (const float*)d_in[1];
  const float* vc = (const float*)d_in[2];
  float* out = (float*)d_out;

  const size_t shmem = (size_t)WAVES * WAVE_LDS * sizeof(float);  // ~264 KB (<320 KB/WGP)
  (void)hipFuncSetAttribute((const void*)gqa_decode_wmma,
                            hipFuncAttributeMaxDynamicSharedMemorySize,
                            (int)shmem);

  dim3 grid(BATCH * NUM_KV);   // 256 workgroups: one per (batch, kv-head)
  dim3 block(WAVES * 32);      // 8 wave32s
  gqa_decode_wmma<<<grid, block, shmem, stream>>>(q, kc, vc, out);
}